// ParticleGNN_83820581749134
// MI455X (gfx1250) — compile-verified
//
#include <hip/hip_runtime.h>
#include <cmath>

#define GRAPHS 64

typedef __attribute__((ext_vector_type(2))) float v2f;
typedef __attribute__((ext_vector_type(8))) float v8f;

__device__ __forceinline__ float gelu_exact(float v) {
  return 0.5f * v * (1.0f + erff(v * 0.7071067811865475f));
}
// order-preserving float<->uint encoding for atomicMax on signed floats
__device__ __forceinline__ unsigned enc_ord(float f) {
  unsigned u = __float_as_uint(f);
  return (u & 0x80000000u) ? ~u : (u | 0x80000000u);
}
__device__ __forceinline__ float dec_ord(unsigned u) {
  return (u & 0x80000000u) ? __uint_as_float(u & 0x7fffffffu)
                           : __uint_as_float(~u);
}

// ---------------- fills ----------------
__global__ void fill_f32(float* p, float v, size_t n) {
  size_t i = (size_t)blockIdx.x * blockDim.x + threadIdx.x;
  if (i < n) p[i] = v;
}
__global__ void fill_u32(unsigned* p, unsigned v, size_t n) {
  size_t i = (size_t)blockIdx.x * blockDim.x + threadIdx.x;
  if (i < n) p[i] = v;
}

// ---------------- GEMM: C[M,64] = act(beta*C + A[M,K] @ W[K,64] + bias) -----
// fp32 WMMA 16x16x4. Block = 256 threads (8 waves) -> 128 rows x 64 cols tile.
// W is staged in LDS pre-swizzled so each lane's B fragment (W[ka][n],
// W[ka+1][n]) is one aligned 8-byte word -> single conflict-free ds_load_b64,
// no VGPR repacking before v_wmma.  act: 0 none, 1 relu, 2 gelu(exact).
__global__ void __launch_bounds__(256)
gemm_n64(const float* __restrict__ A, const float* __restrict__ W,
         const float* __restrict__ bias, float* __restrict__ C,
         int M, int K, int beta, int act)
{
  __shared__ float sW[64 * 64];
  int tid = threadIdx.x;
  for (int i = tid; i < K * 64; i += 256) {
    int k = i >> 6, n = i & 63;
    sW[((k >> 1) * 128) + n * 2 + (k & 1)] = W[i];
  }
  __syncthreads();

  int wave = tid >> 5;
  int lane = tid & 31;
  int half = lane >> 4;
  int lm   = lane & 15;
  int m0   = blockIdx.x * 128 + wave * 16;
  if (m0 >= M) return;  // wave-uniform: EXEC stays all-ones in live waves

  v8f acc[4];
#pragma unroll
  for (int t = 0; t < 4; ++t) {
    if (beta) {
#pragma unroll
      for (int v = 0; v < 8; ++v)
        acc[t][v] = C[(size_t)(m0 + v + 8 * half) * 64 + t * 16 + lm];
    } else {
#pragma unroll
      for (int v = 0; v < 8; ++v) acc[t][v] = 0.0f;
    }
  }

  const float* Arow = A + (size_t)(m0 + lm) * K;
  const v2f*   Bp   = (const v2f*)sW;   // v2f index: (ka>>1)*64 + col
  for (int k0 = 0; k0 < K; k0 += 4) {
    int ka = k0 + 2 * half;             // even; lanes 0-15: K=k0,k0+1 ; 16-31: k0+2,k0+3
    v2f a = *(const v2f*)(Arow + ka);   // aligned 8B global load
    int brow = (ka >> 1) * 64;
#pragma unroll
    for (int t = 0; t < 4; ++t) {
      v2f b = Bp[brow + t * 16 + lm];   // single ds_load_b64, conflict-free
      acc[t] = __builtin_amdgcn_wmma_f32_16x16x4_f32(
          false, a, false, b, (short)0, acc[t], false, false);
    }
  }

#pragma unroll
  for (int t = 0; t < 4; ++t) {
    int col = t * 16 + lm;
    float bb = bias ? bias[col] : 0.0f;
#pragma unroll
    for (int v = 0; v < 8; ++v) {
      float val = acc[t][v] + bb;
      if (act == 1) val = fmaxf(val, 0.0f);
      else if (act == 2) val = gelu_exact(val);
      C[(size_t)(m0 + v + 8 * half) * 64 + col] = val;
    }
  }
}

// ---------------- GAT attention prep: a_s/a_d per (node, head) --------------
__global__ void att_prep(const float* __restrict__ xh, const float* __restrict__ att_src,
                         const float* __restrict__ att_dst, float* aS, float* aD, int N)
{
  size_t t = (size_t)blockIdx.x * blockDim.x + threadIdx.x;
  if (t >= (size_t)N * 4) return;
  int n = (int)(t >> 2), h = (int)(t & 3);
  const float* xr = xh + (size_t)n * 64 + h * 16;
  float s = 0.f, d = 0.f;
#pragma unroll
  for (int i = 0; i < 16; ++i) {
    float v = xr[i];
    s += v * att_src[h * 16 + i];
    d += v * att_dst[h * 16 + i];
  }
  aS[t] = s; aD[t] = d;
}

__device__ __forceinline__ float lrelu02(float a) { return a > 0.f ? a : 0.2f * a; }

// ---------------- GAT softmax pass 1: segmented max over dst ----------------
__global__ void gat_max(const int* __restrict__ ei, const float* __restrict__ aS,
                        const float* __restrict__ aD, unsigned* mxU, int E, int N)
{
  long long e = (long long)blockIdx.x * blockDim.x + threadIdx.x;
  if (e >= (long long)E + N) return;
  int s = (e < E) ? ei[e] : (int)(e - E);
  int d = (e < E) ? ei[(size_t)E + e] : (int)(e - E);
#pragma unroll
  for (int h = 0; h < 4; ++h) {
    float al = lrelu02(aS[(size_t)s * 4 + h] + aD[(size_t)d * 4 + h]);
    atomicMax(&mxU[(size_t)d * 4 + h], enc_ord(al));
  }
}

// ---------------- GAT softmax pass 2: denominator ---------------------------
__global__ void gat_den(const int* __restrict__ ei, const float* __restrict__ aS,
                        const float* __restrict__ aD, const unsigned* __restrict__ mxU,
                        float* den, int E, int N)
{
  long long e = (long long)blockIdx.x * blockDim.x + threadIdx.x;
  if (e >= (long long)E + N) return;
  int s = (e < E) ? ei[e] : (int)(e - E);
  int d = (e < E) ? ei[(size_t)E + e] : (int)(e - E);
#pragma unroll
  for (int h = 0; h < 4; ++h) {
    float al = lrelu02(aS[(size_t)s * 4 + h] + aD[(size_t)d * 4 + h]);
    float ex = expf(al - dec_ord(mxU[(size_t)d * 4 + h]));
    atomicAdd(&den[(size_t)d * 4 + h], ex);
  }
}

// ---------------- GAT accumulate: wave per edge, 2 channels per lane --------
__global__ void gat_accum(const int* __restrict__ ei, const float* __restrict__ xh,
                          const float* __restrict__ aS, const float* __restrict__ aD,
                          const unsigned* __restrict__ mxU, const float* __restrict__ den,
                          float* out, int E, int N)
{
  long long t = (long long)blockIdx.x * blockDim.x + threadIdx.x;
  int lane = (int)(t & 31);
  long long e = t >> 5;
  if (e >= (long long)E + N) return;
  int s = (e < E) ? ei[e] : (int)(e - E);
  int d = (e < E) ? ei[(size_t)E + e] : (int)(e - E);
  int ch = lane * 2;
  int h  = ch >> 4;
  float al = lrelu02(aS[(size_t)s * 4 + h] + aD[(size_t)d * 4 + h]);
  float ex = expf(al - dec_ord(mxU[(size_t)d * 4 + h]));
  float coef = ex / (den[(size_t)d * 4 + h] + 1e-16f);
  const float* xr = xh + (size_t)s * 64 + ch;
  atomicAdd(&out[(size_t)d * 64 + ch + 0], xr[0] * coef);
  atomicAdd(&out[(size_t)d * 64 + ch + 1], xr[1] * coef);
}

// ---------------- degree + inverse sqrt -------------------------------------
__global__ void deg_count(const int* __restrict__ ei, float* deg, int E) {
  size_t e = (size_t)blockIdx.x * blockDim.x + threadIdx.x;
  if (e < (size_t)E) atomicAdd(&deg[ei[(size_t)E + e]], 1.0f);
}
__global__ void deg_invsqrt(float* deg, int N) {
  size_t i = (size_t)blockIdx.x * blockDim.x + threadIdx.x;
  if (i < (size_t)N) { float d = deg[i]; deg[i] = d > 0.f ? 1.0f / sqrtf(d) : 0.0f; }
}

// ---------------- SpMM scatter: out[col] += X[row] * (dis[row]*dis[col]) ----
__global__ void spmm_scatter(const int* __restrict__ ei, const float* __restrict__ X,
                             const float* __restrict__ dis, float* out, int E)
{
  long long t = (long long)blockIdx.x * blockDim.x + threadIdx.x;
  int lane = (int)(t & 31);
  long long e = t >> 5;
  if (e >= (long long)E) return;
  int r = ei[e];
  int c = ei[(size_t)E + e];
  float w = dis ? dis[r] * dis[c] : 1.0f;
  int ch = lane * 2;
  const float* xr = X + (size_t)r * 64 + ch;
  atomicAdd(&out[(size_t)c * 64 + ch + 0], xr[0] * w);
  atomicAdd(&out[(size_t)c * 64 + ch + 1], xr[1] * w);
}

// ---------------- per-graph node counts (batch is sorted) -------------------
__global__ void seg_cnt(const int* __restrict__ batch, int N, float* cnt) {
  int t = blockIdx.x * blockDim.x + threadIdx.x;   // 2048 threads total
  int per = (N + 2047) / 2048;
  int s = t * per, e = min(s + per, N);
  if (s >= e) return;
  int g = batch[s]; float acc = 0.f;
  for (int i = s; i < e; ++i) {
    int gi = batch[i];
    if (gi != g) { atomicAdd(&cnt[g], acc); acc = 0.f; g = gi; }
    acc += 1.0f;
  }
  atomicAdd(&cnt[g], acc);
}

// ---------------- segmented channel sums (64 threads = channels) ------------
#define SEG_CHUNK 512
__global__ void seg_sum64(const float* __restrict__ V, const float* __restrict__ bias,
                          const int* __restrict__ batch, int N, float* gsum)
{
  int c = threadIdx.x;
  int s = blockIdx.x * SEG_CHUNK;
  int e = min(s + SEG_CHUNK, N);
  if (s >= e) return;
  float bb = bias ? bias[c] : 0.0f;
  float acc = 0.f;
  int g = batch[s];
  for (int i = s; i < e; ++i) {
    int gi = batch[i];
    if (gi != g) { atomicAdd(&gsum[(size_t)g * 64 + c], acc); acc = 0.f; g = gi; }
    acc += V[(size_t)i * 64 + c] + bb;
  }
  atomicAdd(&gsum[(size_t)g * 64 + c], acc);
}

__global__ void seg_var64(const float* __restrict__ V, const float* __restrict__ bias,
                          const int* __restrict__ batch, int N,
                          const float* __restrict__ gsum, const float* __restrict__ cnt,
                          const float* __restrict__ ms, float* gvar)
{
  int c = threadIdx.x;
  int s = blockIdx.x * SEG_CHUNK;
  int e = min(s + SEG_CHUNK, N);
  if (s >= e) return;
  float bb = bias ? bias[c] : 0.0f;
  float m  = ms[c];
  int g = batch[s];
  float mean = gsum[(size_t)g * 64 + c] / fmaxf(cnt[g], 1.0f);
  float acc = 0.f;
  for (int i = s; i < e; ++i) {
    int gi = batch[i];
    if (gi != g) {
      atomicAdd(&gvar[(size_t)g * 64 + c], acc);
      acc = 0.f; g = gi;
      mean = gsum[(size_t)g * 64 + c] / fmaxf(cnt[g], 1.0f);
    }
    float dd = V[(size_t)i * 64 + c] + bb - m * mean;
    acc += dd * dd;
  }
  atomicAdd(&gvar[(size_t)g * 64 + c], acc);
}

// out = relu( w * (v + bias - ms*mean) * rsqrt(var + eps) + b )
__global__ void norm_apply_relu(const float* __restrict__ V, const float* __restrict__ bias,
                                const int* __restrict__ batch,
                                const float* __restrict__ gsum, const float* __restrict__ gvar,
                                const float* __restrict__ cnt,
                                const float* __restrict__ w, const float* __restrict__ b,
                                const float* __restrict__ ms, float* out, int N)
{
  size_t idx = (size_t)blockIdx.x * blockDim.x + threadIdx.x;
  if (idx >= (size_t)N * 64) return;
  int i = (int)(idx >> 6);
  int c = (int)(idx & 63);
  int g = batch[i];
  float cn   = fmaxf(cnt[g], 1.0f);
  float mean = gsum[(size_t)g * 64 + c] / cn;
  float var  = gvar[(size_t)g * 64 + c] / cn;
  float d = V[idx] + (bias ? bias[c] : 0.0f) - ms[c] * mean;
  float r = w[c] * d * rsqrtf(var + 1e-5f) + b[c];
  out[idx] = fmaxf(r, 0.0f);
}

// ---------------- global max + sum pool (values >= 0 after relu) ------------
__global__ void seg_pool64(const float* __restrict__ V, const int* __restrict__ batch,
                           int N, float* psum, unsigned* pmax)
{
  int c = threadIdx.x;
  int s = blockIdx.x * SEG_CHUNK;
  int e = min(s + SEG_CHUNK, N);
  if (s >= e) return;
  float accS = 0.f, accM = 0.f;
  int g = batch[s];
  for (int i = s; i < e; ++i) {
    int gi = batch[i];
    if (gi != g) {
      atomicAdd(&psum[(size_t)g * 64 + c], accS);
      atomicMax(&pmax[(size_t)g * 64 + c], __float_as_uint(accM));
      accS = 0.f; accM = 0.f; g = gi;
    }
    float v = V[(size_t)i * 64 + c];
    accS += v;
    accM = fmaxf(accM, v);
  }
  atomicAdd(&psum[(size_t)g * 64 + c], accS);
  atomicMax(&pmax[(size_t)g * 64 + c], __float_as_uint(accM));
}

// ---------------- classifier: [G,128] -> GELU 64 -> 2 -> log_softmax --------
__global__ void classifier(const float* __restrict__ pmax, const float* __restrict__ psum,
                           const float* __restrict__ cnt,
                           const float* __restrict__ w1, const float* __restrict__ b1,
                           const float* __restrict__ w2, const float* __restrict__ b2,
                           float* out)
{
  int g = threadIdx.x;
  if (g >= GRAPHS) return;
  float cn = fmaxf(cnt[g], 1.0f);
  float l0 = b2[0], l1 = b2[1];
  for (int j = 0; j < 64; ++j) {
    float a = b1[j];
    for (int i = 0; i < 128; ++i) {
      float zi = (i < 64) ? pmax[(size_t)g * 64 + i]
                          : psum[(size_t)g * 64 + (i - 64)] / cn;
      a += zi * w1[(size_t)i * 64 + j];
    }
    a = gelu_exact(a);
    l0 += a * w2[j * 2 + 0];
    l1 += a * w2[j * 2 + 1];
  }
  float m = fmaxf(l0, l1);
  float lse = m + logf(expf(l0 - m) + expf(l1 - m));
  out[g * 2 + 0] = l0 - lse;
  out[g * 2 + 1] = l1 - lse;
}

// ============================================================================
extern "C" void kernel_launch(void* const* d_in, const int* in_sizes, int n_in,
                              void* d_out, int out_size, void* d_ws, size_t ws_size,
                              hipStream_t stream)
{
  const float* xin      = (const float*)d_in[0];   // [N,8]
  const int*   ei       = (const int*)  d_in[1];   // [2,E]
  const int*   batch    = (const int*)  d_in[2];   // [N]
  const float* enc_w    = (const float*)d_in[3];
  const float* enc_b    = (const float*)d_in[4];
  const float* gat_w    = (const float*)d_in[5];
  const float* att_src  = (const float*)d_in[6];
  const float* att_dst  = (const float*)d_in[7];
  const float* gat_b    = (const float*)d_in[8];
  const float* tag_w    = (const float*)d_in[9];   // [4,64,64]
  const float* tag_b    = (const float*)d_in[10];
  const float* gc_w_rel = (const float*)d_in[11];
  const float* gc_b_rel = (const float*)d_in[12];
  const float* gc_w_root= (const float*)d_in[13];
  const float* n1_w = (const float*)d_in[14];
  const float* n1_b = (const float*)d_in[15];
  const float* n1_ms= (const float*)d_in[16];
  const float* n2_w = (const float*)d_in[17];
  const float* n2_b = (const float*)d_in[18];
  const float* n2_ms= (const float*)d_in[19];
  const float* cls_w1 = (const float*)d_in[20];
  const float* cls_b1 = (const float*)d_in[21];
  const float* cls_w2 = (const float*)d_in[22];
  const float* cls_b2 = (const float*)d_in[23];

  const int N = in_sizes[0] / 8;
  const int E = in_sizes[1] / 2;
  const size_t NH = (size_t)N * 64;

  // ---- workspace layout (floats) ----
  float*    XA   = (float*)d_ws;
  float*    XB   = XA + NH;
  float*    XC   = XB + NH;
  float*    attS = XC + NH;
  float*    attD = attS + (size_t)N * 4;
  float*    den  = attD + (size_t)N * 4;
  unsigned* mxU  = (unsigned*)(den + (size_t)N * 4);
  float*    deg  = (float*)(mxU + (size_t)N * 4);
  float*    cnt  = deg + N;
  float*    gsum = cnt + GRAPHS;
  float*    gvar = gsum + (size_t)GRAPHS * 64;
  float*    psum = gvar + (size_t)GRAPHS * 64;
  unsigned* pmax = (unsigned*)(psum + (size_t)GRAPHS * 64);

  auto cdiv = [](long long a, long long b) { return (unsigned)((a + b - 1) / b); };
  const unsigned gElem = cdiv(NH, 256);
  const unsigned gRows = cdiv(N, 128);
  const unsigned gEdge = cdiv((long long)E + N, 256);
  const unsigned gEdgeW = cdiv(((long long)E + N) * 32, 256);
  const unsigned gEw    = cdiv((long long)E * 32, 256);
  const unsigned gSeg   = cdiv(N, SEG_CHUNK);

  // ---- 1. encoder: XA = gelu(x @ enc_w + enc_b) ----
  gemm_n64<<<gRows, 256, 0, stream>>>(xin, enc_w, enc_b, XA, N, 8, 0, 2);

  // ---- 2. GAT: xh = XA @ gat_w -> XB ----
  gemm_n64<<<gRows, 256, 0, stream>>>(XA, gat_w, nullptr, XB, N, 64, 0, 0);
  att_prep<<<cdiv((long long)N * 4, 256), 256, 0, stream>>>(XB, att_src, att_dst, attS, attD, N);

  fill_u32<<<cdiv((long long)N * 4, 256), 256, 0, stream>>>(mxU, 0x00800000u /*enc(-FLT_MAX)*/, (size_t)N * 4);
  fill_f32<<<cdiv((long long)N * 4, 256), 256, 0, stream>>>(den, 0.0f, (size_t)N * 4);
  fill_f32<<<gElem, 256, 0, stream>>>(XA, 0.0f, NH);

  gat_max  <<<gEdge,  256, 0, stream>>>(ei, attS, attD, mxU, E, N);
  gat_den  <<<gEdge,  256, 0, stream>>>(ei, attS, attD, mxU, den, E, N);
  gat_accum<<<gEdgeW, 256, 0, stream>>>(ei, XB, attS, attD, mxU, den, XA, E, N);

  // ---- graph norm 1 (+gat_b) + relu -> XB ----
  fill_f32<<<1, 256, 0, stream>>>(cnt, 0.0f, GRAPHS);
  seg_cnt<<<32, 64, 0, stream>>>(batch, N, cnt);
  fill_f32<<<cdiv(GRAPHS * 64, 256), 256, 0, stream>>>(gsum, 0.0f, (size_t)GRAPHS * 64);
  fill_f32<<<cdiv(GRAPHS * 64, 256), 256, 0, stream>>>(gvar, 0.0f, (size_t)GRAPHS * 64);
  seg_sum64<<<gSeg, 64, 0, stream>>>(XA, gat_b, batch, N, gsum);
  seg_var64<<<gSeg, 64, 0, stream>>>(XA, gat_b, batch, N, gsum, cnt, n1_ms, gvar);
  norm_apply_relu<<<gElem, 256, 0, stream>>>(XA, gat_b, batch, gsum, gvar, cnt,
                                             n1_w, n1_b, n1_ms, XB, N);

  // ---- 3. TAGConv ----
  fill_f32<<<cdiv(N, 256), 256, 0, stream>>>(deg, 0.0f, (size_t)N);
  deg_count  <<<cdiv(E, 256), 256, 0, stream>>>(ei, deg, E);
  deg_invsqrt<<<cdiv(N, 256), 256, 0, stream>>>(deg, N);

  gemm_n64<<<gRows, 256, 0, stream>>>(XB, tag_w + 0 * 4096, nullptr, XC, N, 64, 0, 0);
  fill_f32<<<gElem, 256, 0, stream>>>(XA, 0.0f, NH);
  spmm_scatter<<<gEw, 256, 0, stream>>>(ei, XB, deg, XA, E);
  gemm_n64<<<gRows, 256, 0, stream>>>(XA, tag_w + 1 * 4096, nullptr, XC, N, 64, 1, 0);
  fill_f32<<<gElem, 256, 0, stream>>>(XB, 0.0f, NH);
  spmm_scatter<<<gEw, 256, 0, stream>>>(ei, XA, deg, XB, E);
  gemm_n64<<<gRows, 256, 0, stream>>>(XB, tag_w + 2 * 4096, nullptr, XC, N, 64, 1, 0);
  fill_f32<<<gElem, 256, 0, stream>>>(XA, 0.0f, NH);
  spmm_scatter<<<gEw, 256, 0, stream>>>(ei, XB, deg, XA, E);
  gemm_n64<<<gRows, 256, 0, stream>>>(XA, tag_w + 3 * 4096, nullptr, XC, N, 64, 1, 0);

  // ---- graph norm 2 (+tag_b) + relu -> XB ----
  fill_f32<<<cdiv(GRAPHS * 64, 256), 256, 0, stream>>>(gsum, 0.0f, (size_t)GRAPHS * 64);
  fill_f32<<<cdiv(GRAPHS * 64, 256), 256, 0, stream>>>(gvar, 0.0f, (size_t)GRAPHS * 64);
  seg_sum64<<<gSeg, 64, 0, stream>>>(XC, tag_b, batch, N, gsum);
  seg_var64<<<gSeg, 64, 0, stream>>>(XC, tag_b, batch, N, gsum, cnt, n2_ms, gvar);
  norm_apply_relu<<<gElem, 256, 0, stream>>>(XC, tag_b, batch, gsum, gvar, cnt,
                                             n2_w, n2_b, n2_ms, XB, N);

  // ---- 4. GraphConv: XC = relu(agg@W_rel + b_rel + XB@W_root) ----
  fill_f32<<<gElem, 256, 0, stream>>>(XA, 0.0f, NH);
  spmm_scatter<<<gEw, 256, 0, stream>>>(ei, XB, nullptr, XA, E);
  gemm_n64<<<gRows, 256, 0, stream>>>(XA, gc_w_rel, gc_b_rel, XC, N, 64, 0, 0);
  gemm_n64<<<gRows, 256, 0, stream>>>(XB, gc_w_root, nullptr, XC, N, 64, 1, 1);

  // ---- 5. pool + classifier ----
  fill_f32<<<cdiv(GRAPHS * 64, 256), 256, 0, stream>>>(psum, 0.0f, (size_t)GRAPHS * 64);
  fill_u32<<<cdiv(GRAPHS * 64, 256), 256, 0, stream>>>(pmax, 0u, (size_t)GRAPHS * 64);
  seg_pool64<<<gSeg, 64, 0, stream>>>(XC, batch, N, psum, pmax);

  classifier<<<1, 64, 0, stream>>>((const float*)pmax, psum, cnt,
                                   cls_w1, cls_b1, cls_w2, cls_b2, (float*)d_out);
}